// CausalSelfAttention_34574486733066
// MI455X (gfx1250) — compile-verified
//
#include <hip/hip_runtime.h>
#include <math.h>

// ---------------------------------------------------------------------------
// Causal self-attention fwd for MI455X (gfx1250), fp32 via V_WMMA_F32_16X16X4_F32
// ---------------------------------------------------------------------------

typedef float v2f __attribute__((ext_vector_type(2)));
typedef float v8f __attribute__((ext_vector_type(8)));

#define N_HEADS  8
#define HEAD_DIM 64
#define DMODEL   512
#define SEQ      4096
#define BATCH    2

__device__ __forceinline__ v8f wmma_f32(v2f a, v2f b, v8f c) {
  // D = A(16x4 f32) * B(4x16 f32) + C(16x16 f32), wave32
  return __builtin_amdgcn_wmma_f32_16x16x4_f32(false, a, false, b, (short)0, c,
                                               false, false);
}

// ---------------------------------------------------------------------------
// Kernel 1: QKV projection (out = x @ W^T) with fused RoPE epilogue.
// grid = (M/128, 12); by selects weight (by>>2) and n-block ((by&3)*128).
// Block = 256 threads = 8 waves; wave (wm,wn) computes a 32x64 sub-tile.
// ---------------------------------------------------------------------------
__global__ __launch_bounds__(256, 1)
void qkv_rope_kernel(const float* __restrict__ X,
                     const float* __restrict__ Wq,
                     const float* __restrict__ Wk,
                     const float* __restrict__ Wv,
                     const float* __restrict__ Cos,
                     const float* __restrict__ Sin,
                     float* __restrict__ Qo,
                     float* __restrict__ Ko,
                     float* __restrict__ Vo) {
  __shared__ float As[128 * 34];   // x tile   [128 rows][32 k] (+2 pad)
  __shared__ float Bs[128 * 34];   // W tile   [128 n   ][32 k] (+2 pad)

  const int tid  = threadIdx.x;
  const int lane = tid & 31;
  const int wid  = tid >> 5;
  const int wm   = wid & 3;        // 32-row strip within 128
  const int wn   = wid >> 2;       // 64-col strip within 128
  const int lhi  = lane >> 4;      // 0/1 : K-half for A/B frags
  const int llo  = lane & 15;

  const int m0   = blockIdx.x * 128;
  const int by   = blockIdx.y;
  const int wsel = by >> 2;                 // 0=Q 1=K 2=V
  const int n0   = (by & 3) * 128;
  const float* W = (wsel == 0) ? Wq : (wsel == 1) ? Wk : Wv;

  v8f acc[2][4];
#pragma unroll
  for (int i = 0; i < 2; ++i)
#pragma unroll
    for (int j = 0; j < 4; ++j) acc[i][j] = {};

  for (int kt = 0; kt < 16; ++kt) {
    const int k0 = kt * 32;
    // cooperative fill: 2048 float2 per matrix -> 8 per thread
#pragma unroll
    for (int i = 0; i < 8; ++i) {
      const int linear = tid + i * 256;     // 0..2047
      const int r  = linear >> 4;           // row 0..127
      const int c2 = (linear & 15) * 2;     // even float col
      *(v2f*)&As[r * 34 + c2] =
          *(const v2f*)&X[(size_t)(m0 + r) * DMODEL + k0 + c2];
      *(v2f*)&Bs[r * 34 + c2] =
          *(const v2f*)&W[(size_t)(n0 + r) * DMODEL + k0 + c2];
    }
    __syncthreads();

#pragma unroll
    for (int kk = 0; kk < 8; ++kk) {
      const int kc = kk * 4 + lhi * 2;
      v2f a0 = *(const v2f*)&As[(wm * 32 + 0 * 16 + llo) * 34 + kc];
      v2f a1 = *(const v2f*)&As[(wm * 32 + 1 * 16 + llo) * 34 + kc];
      v2f b0 = *(const v2f*)&Bs[(wn * 64 + 0 * 16 + llo) * 34 + kc];
      v2f b1 = *(const v2f*)&Bs[(wn * 64 + 1 * 16 + llo) * 34 + kc];
      v2f b2 = *(const v2f*)&Bs[(wn * 64 + 2 * 16 + llo) * 34 + kc];
      v2f b3 = *(const v2f*)&Bs[(wn * 64 + 3 * 16 + llo) * 34 + kc];
      acc[0][0] = wmma_f32(a0, b0, acc[0][0]);
      acc[0][1] = wmma_f32(a0, b1, acc[0][1]);
      acc[0][2] = wmma_f32(a0, b2, acc[0][2]);
      acc[0][3] = wmma_f32(a0, b3, acc[0][3]);
      acc[1][0] = wmma_f32(a1, b0, acc[1][0]);
      acc[1][1] = wmma_f32(a1, b1, acc[1][1]);
      acc[1][2] = wmma_f32(a1, b2, acc[1][2]);
      acc[1][3] = wmma_f32(a1, b3, acc[1][3]);
    }
    __syncthreads();
  }

  // ---- epilogue: scatter to [B][H][T][64]; RoPE for Q/K -------------------
  if (wsel == 2) {
#pragma unroll
    for (int mi = 0; mi < 2; ++mi)
#pragma unroll
      for (int r = 0; r < 8; ++r) {
        const int row = m0 + wm * 32 + mi * 16 + lhi * 8 + r;
        const int bb = row >> 12, t = row & (SEQ - 1);
#pragma unroll
        for (int ni = 0; ni < 4; ++ni) {
          const int col = n0 + wn * 64 + ni * 16 + llo;
          const int h = col >> 6, d = col & 63;
          Vo[(((size_t)bb * N_HEADS + h) * SEQ + t) * HEAD_DIM + d] =
              acc[mi][ni][r];
        }
      }
  } else {
    float* Out = (wsel == 0) ? Qo : Ko;
    const int h = (n0 + wn * 64) >> 6;       // wave spans exactly one head
#pragma unroll
    for (int mi = 0; mi < 2; ++mi)
#pragma unroll
      for (int r = 0; r < 8; ++r) {
        const int row = m0 + wm * 32 + mi * 16 + lhi * 8 + r;
        const int bb = row >> 12, t = row & (SEQ - 1);
        const size_t base = (((size_t)bb * N_HEADS + h) * SEQ + t) * HEAD_DIM;
#pragma unroll
        for (int ni = 0; ni < 2; ++ni) {     // dd in [0,32)
          const int dd = ni * 16 + llo;
          const float c = Cos[t * 32 + dd];
          const float s = Sin[t * 32 + dd];
          const float x1 = acc[mi][ni][r];        // d = dd
          const float x2 = acc[mi][ni + 2][r];    // d = dd + 32
          Out[base + dd]      =  x1 * c + x2 * s;
          Out[base + dd + 32] = -x1 * s + x2 * c;
        }
      }
  }
}

// ---------------------------------------------------------------------------
// Kernel 2: flash attention. grid = (T/64, B*H), block = 128 (4 waves).
// Wave w owns 16 query rows; Q frags live in registers (pre-scaled by 1/8).
// ---------------------------------------------------------------------------
__global__ __launch_bounds__(128, 1)
void flash_kernel(const float* __restrict__ Q, const float* __restrict__ K,
                  const float* __restrict__ V, float* __restrict__ Y) {
  __shared__ float Ks[64 * 66];
  __shared__ float Vs[64 * 66];
  __shared__ float Ps[4][16 * 66];

  const int tid  = threadIdx.x;
  const int lane = tid & 31;
  const int wv   = tid >> 5;
  const int lhi  = lane >> 4, llo = lane & 15;
  const int qb   = blockIdx.x;
  const int bh   = blockIdx.y;
  const int q0   = qb * 64;

  const float* Qb = Q + (size_t)bh * SEQ * HEAD_DIM;
  const float* Kb = K + (size_t)bh * SEQ * HEAD_DIM;
  const float* Vb = V + (size_t)bh * SEQ * HEAD_DIM;

  // Q fragments: A-layout, contraction over d=64 (16 k-steps of 4)
  v2f qf[16];
  {
    const int qrow = q0 + wv * 16 + llo;
#pragma unroll
    for (int kk = 0; kk < 16; ++kk) {
      qf[kk] = *(const v2f*)&Qb[(size_t)qrow * HEAD_DIM + kk * 4 + lhi * 2];
      qf[kk] *= 0.125f;   // 1/sqrt(64)
    }
  }

  float m_i[8], l_i[8];
#pragma unroll
  for (int r = 0; r < 8; ++r) { m_i[r] = -INFINITY; l_i[r] = 0.f; }
  v8f oacc[4];
#pragma unroll
  for (int ni = 0; ni < 4; ++ni) oacc[ni] = {};

  for (int kb = 0; kb <= qb; ++kb) {
    const int k0 = kb * 64;
    // stage K,V tiles: 2048 float2 each, 16 per thread
#pragma unroll
    for (int i = 0; i < 16; ++i) {
      const int linear = tid + i * 128;
      const int r  = linear >> 5;
      const int c2 = (linear & 31) * 2;
      *(v2f*)&Ks[r * 66 + c2] = *(const v2f*)&Kb[(size_t)(k0 + r) * HEAD_DIM + c2];
      *(v2f*)&Vs[r * 66 + c2] = *(const v2f*)&Vb[(size_t)(k0 + r) * HEAD_DIM + c2];
    }
    __syncthreads();

    // S = q . k^T  (16x64 per wave)
    v8f sacc[4];
#pragma unroll
    for (int ni = 0; ni < 4; ++ni) sacc[ni] = {};
#pragma unroll
    for (int kk = 0; kk < 16; ++kk) {
      const int kc = kk * 4 + lhi * 2;
#pragma unroll
      for (int ni = 0; ni < 4; ++ni) {
        v2f b = *(const v2f*)&Ks[(ni * 16 + llo) * 66 + kc];
        sacc[ni] = wmma_f32(qf[kk], b, sacc[ni]);
      }
    }

    // causal mask on diagonal block (uniform branch)
    if (kb == qb) {
#pragma unroll
      for (int ni = 0; ni < 4; ++ni) {
        const int ncol = ni * 16 + llo;
#pragma unroll
        for (int r = 0; r < 8; ++r) {
          const int mrow = wv * 16 + lhi * 8 + r;
          if (ncol > mrow) sacc[ni][r] = -INFINITY;
        }
      }
    }

    // online softmax: rows live across 16 lanes of each half-wave
#pragma unroll
    for (int r = 0; r < 8; ++r) {
      float mx = fmaxf(fmaxf(sacc[0][r], sacc[1][r]),
                       fmaxf(sacc[2][r], sacc[3][r]));
#pragma unroll
      for (int off = 1; off < 16; off <<= 1)
        mx = fmaxf(mx, __shfl_xor(mx, off, 32));
      const float mnew  = fmaxf(m_i[r], mx);
      const float alpha = __expf(m_i[r] - mnew);
      float sum = 0.f;
#pragma unroll
      for (int ni = 0; ni < 4; ++ni) {
        const float p = __expf(sacc[ni][r] - mnew);
        sacc[ni][r] = p;
        sum += p;
      }
#pragma unroll
      for (int off = 1; off < 16; off <<= 1)
        sum += __shfl_xor(sum, off, 32);
      m_i[r] = mnew;
      l_i[r] = l_i[r] * alpha + sum;
#pragma unroll
      for (int ni = 0; ni < 4; ++ni) oacc[ni][r] *= alpha;
    }

    // P (D-layout) -> LDS (per-wave, same-wave in-order, no barrier needed)
#pragma unroll
    for (int ni = 0; ni < 4; ++ni)
#pragma unroll
      for (int r = 0; r < 8; ++r)
        Ps[wv][(lhi * 8 + r) * 66 + ni * 16 + llo] = sacc[ni][r];

    // O += P @ V  (contraction over kt=64)
#pragma unroll
    for (int kk = 0; kk < 16; ++kk) {
      const int kc = kk * 4 + lhi * 2;
      v2f a = *(const v2f*)&Ps[wv][llo * 66 + kc];
#pragma unroll
      for (int ni = 0; ni < 4; ++ni) {
        v2f b;
        b.x = Vs[(kc + 0) * 66 + ni * 16 + llo];
        b.y = Vs[(kc + 1) * 66 + ni * 16 + llo];
        oacc[ni] = wmma_f32(a, b, oacc[ni]);
      }
    }
    __syncthreads();   // protect Ks/Vs before next stage
  }

  // normalize and write y as [B*T][512] (col = h*64 + d)
  const int bb = bh >> 3, h = bh & 7;
#pragma unroll
  for (int ni = 0; ni < 4; ++ni)
#pragma unroll
    for (int r = 0; r < 8; ++r) {
      const int t   = q0 + wv * 16 + lhi * 8 + r;
      const int col = h * 64 + ni * 16 + llo;
      Y[((size_t)bb * SEQ + t) * DMODEL + col] = oacc[ni][r] / l_i[r];
    }
}

// ---------------------------------------------------------------------------
// Kernel 3: output projection out = y @ Wp^T. grid = (64, 4), block 256.
// ---------------------------------------------------------------------------
__global__ __launch_bounds__(256, 1)
void proj_kernel(const float* __restrict__ X, const float* __restrict__ W,
                 float* __restrict__ Out) {
  __shared__ float As[128 * 34];
  __shared__ float Bs[128 * 34];

  const int tid  = threadIdx.x;
  const int lane = tid & 31;
  const int wid  = tid >> 5;
  const int wm   = wid & 3;
  const int wn   = wid >> 2;
  const int lhi  = lane >> 4, llo = lane & 15;
  const int m0   = blockIdx.x * 128;
  const int n0   = blockIdx.y * 128;

  v8f acc[2][4];
#pragma unroll
  for (int i = 0; i < 2; ++i)
#pragma unroll
    for (int j = 0; j < 4; ++j) acc[i][j] = {};

  for (int kt = 0; kt < 16; ++kt) {
    const int k0 = kt * 32;
#pragma unroll
    for (int i = 0; i < 8; ++i) {
      const int linear = tid + i * 256;
      const int r  = linear >> 4;
      const int c2 = (linear & 15) * 2;
      *(v2f*)&As[r * 34 + c2] =
          *(const v2f*)&X[(size_t)(m0 + r) * DMODEL + k0 + c2];
      *(v2f*)&Bs[r * 34 + c2] =
          *(const v2f*)&W[(size_t)(n0 + r) * DMODEL + k0 + c2];
    }
    __syncthreads();
#pragma unroll
    for (int kk = 0; kk < 8; ++kk) {
      const int kc = kk * 4 + lhi * 2;
      v2f a0 = *(const v2f*)&As[(wm * 32 + 0 * 16 + llo) * 34 + kc];
      v2f a1 = *(const v2f*)&As[(wm * 32 + 1 * 16 + llo) * 34 + kc];
      v2f b0 = *(const v2f*)&Bs[(wn * 64 + 0 * 16 + llo) * 34 + kc];
      v2f b1 = *(const v2f*)&Bs[(wn * 64 + 1 * 16 + llo) * 34 + kc];
      v2f b2 = *(const v2f*)&Bs[(wn * 64 + 2 * 16 + llo) * 34 + kc];
      v2f b3 = *(const v2f*)&Bs[(wn * 64 + 3 * 16 + llo) * 34 + kc];
      acc[0][0] = wmma_f32(a0, b0, acc[0][0]);
      acc[0][1] = wmma_f32(a0, b1, acc[0][1]);
      acc[0][2] = wmma_f32(a0, b2, acc[0][2]);
      acc[0][3] = wmma_f32(a0, b3, acc[0][3]);
      acc[1][0] = wmma_f32(a1, b0, acc[1][0]);
      acc[1][1] = wmma_f32(a1, b1, acc[1][1]);
      acc[1][2] = wmma_f32(a1, b2, acc[1][2]);
      acc[1][3] = wmma_f32(a1, b3, acc[1][3]);
    }
    __syncthreads();
  }

#pragma unroll
  for (int mi = 0; mi < 2; ++mi)
#pragma unroll
    for (int r = 0; r < 8; ++r) {
      const int row = m0 + wm * 32 + mi * 16 + lhi * 8 + r;
#pragma unroll
      for (int ni = 0; ni < 4; ++ni) {
        const int col = n0 + wn * 64 + ni * 16 + llo;
        Out[(size_t)row * DMODEL + col] = acc[mi][ni][r];
      }
    }
}

// ---------------------------------------------------------------------------
extern "C" void kernel_launch(void* const* d_in, const int* in_sizes, int n_in,
                              void* d_out, int out_size, void* d_ws,
                              size_t ws_size, hipStream_t stream) {
  const float* x    = (const float*)d_in[0];
  const float* cosp = (const float*)d_in[1];
  const float* sinp = (const float*)d_in[2];
  const float* Wq   = (const float*)d_in[3];
  const float* Wk   = (const float*)d_in[4];
  const float* Wv   = (const float*)d_in[5];
  const float* Wp   = (const float*)d_in[6];
  float* out = (float*)d_out;

  const size_t elems = (size_t)BATCH * N_HEADS * SEQ * HEAD_DIM;  // 4M floats
  float* qbuf = (float*)d_ws;
  float* kbuf = qbuf + elems;
  float* vbuf = kbuf + elems;
  float* ybuf = vbuf + elems;

  dim3 gq(64, 12);
  qkv_rope_kernel<<<gq, 256, 0, stream>>>(x, Wq, Wk, Wv, cosp, sinp,
                                          qbuf, kbuf, vbuf);
  dim3 gf(64, 16);
  flash_kernel<<<gf, 128, 0, stream>>>(qbuf, kbuf, vbuf, ybuf);
  dim3 gp(64, 4);
  proj_kernel<<<gp, 256, 0, stream>>>(ybuf, Wp, out);
}